// TopKRouter_12455405158652
// MI455X (gfx1250) — compile-verified
//
#include <hip/hip_runtime.h>

typedef float v2f __attribute__((ext_vector_type(2)));
typedef float v8f __attribute__((ext_vector_type(8)));

#define NUM_EXPERTS 8
#define TOKENS_PER_BLOCK 128
#define THREADS 256

// Zero the 8-float tokens_per_expert region (d_out is poisoned by the harness;
// must be deterministic per call, so zero every launch on the same stream).
__global__ void router_zero_counts(float* __restrict__ counts) {
    if (threadIdx.x < NUM_EXPERTS) counts[threadIdx.x] = 0.0f;
}

__global__ __launch_bounds__(THREADS)
void topk_router_wmma(const float* __restrict__ x,
                      const float* __restrict__ W,
                      float* __restrict__ scores,     // [T,2] fp32
                      float* __restrict__ idx_out,    // [T,2] indices (as float)
                      float* __restrict__ counts,     // [8] fp32, pre-zeroed
                      int H) {
    __shared__ float logitsLds[TOKENS_PER_BLOCK][NUM_EXPERTS];
    __shared__ int   hist[NUM_EXPERTS];

    const int tid  = threadIdx.x;
    if (tid < NUM_EXPERTS) hist[tid] = 0;

    const int wave = tid >> 5;          // 8 waves of 32 (wave32)
    const int lane = tid & 31;
    const int h    = lane >> 4;         // lane half: selects K slots {0,1} vs {2,3}
    const int mn   = lane & 15;         // A: row M; B: col N (expert)

    const long tok0 = (long)blockIdx.x * TOKENS_PER_BLOCK + wave * 16;
    const float* xRow = x + (tok0 + mn) * (long)H;
    // Experts 8..15 duplicate rows 0..7 -> garbage logit columns we never store.
    const float* wRow = W + (long)(mn & 7) * (long)H;

    // Each iteration covers K=8 physical columns with one B128 per lane per
    // matrix and two V_WMMA_F32_16X16X4_F32 ops. Physical column -> hardware
    // K-slot remap (identical for A and B, so the K-sum is unchanged):
    //   wmma#0: half0 slots K0,K1 = cols 8i+0,1 ; half1 slots K2,K3 = cols 8i+4,5
    //   wmma#1: half0 slots K0,K1 = cols 8i+2,3 ; half1 slots K2,K3 = cols 8i+6,7
    v8f acc = {};
    const int iters = H >> 3;
    #pragma unroll 4
    for (int i = 0; i < iters; ++i) {
        const int c0 = (i << 3) + (h << 2);
        const float4 xa = *reinterpret_cast<const float4*>(xRow + c0);
        const float4 wb = *reinterpret_cast<const float4*>(wRow + c0);
        v2f a0 = {xa.x, xa.y}, b0 = {wb.x, wb.y};
        v2f a1 = {xa.z, xa.w}, b1 = {wb.z, wb.w};
        acc = __builtin_amdgcn_wmma_f32_16x16x4_f32(
            false, a0, false, b0, (short)0, acc, false, false);
        acc = __builtin_amdgcn_wmma_f32_16x16x4_f32(
            false, a1, false, b1, (short)0, acc, false, false);
    }

    // C/D layout: acc[r] at this lane = logits[token tok0 + r + 8*h][expert mn].
    if (mn < NUM_EXPERTS) {
        #pragma unroll
        for (int r = 0; r < 8; ++r)
            logitsLds[wave * 16 + r + 8 * h][mn] = acc[r];
    }
    __syncthreads();

    if (tid < TOKENS_PER_BLOCK) {
        float l[NUM_EXPERTS];
        #pragma unroll
        for (int e = 0; e < NUM_EXPERTS; ++e) l[e] = logitsLds[tid][e];

        // top-2, descending, ties -> lower index first (matches jax.lax.top_k)
        float best = l[0]; int bi = 0;
        float sec  = -__builtin_inff(); int si = 0;
        #pragma unroll
        for (int e = 1; e < NUM_EXPERTS; ++e) {
            const float v = l[e];
            if (v > best)      { sec = best; si = bi; best = v; bi = e; }
            else if (v > sec)  { sec = v; si = e; }
        }

        // fp32 softmax over the two selected logits
        const float e1 = expf(sec - best);
        const float inv = 1.0f / (1.0f + e1);
        const float s0 = inv;
        const float s1 = e1 * inv;

        const long gt = (long)blockIdx.x * TOKENS_PER_BLOCK + tid;
        scores[2 * gt]      = s0;
        scores[2 * gt + 1]  = s1;
        idx_out[2 * gt]     = (float)bi;
        idx_out[2 * gt + 1] = (float)si;

        atomicAdd(&hist[bi], 1);
        atomicAdd(&hist[si], 1);
    }
    __syncthreads();

    if (tid < NUM_EXPERTS)
        atomicAdd(&counts[tid], (float)hist[tid]);
}

extern "C" void kernel_launch(void* const* d_in, const int* in_sizes, int n_in,
                              void* d_out, int out_size, void* d_ws, size_t ws_size,
                              hipStream_t stream) {
    const float* x = (const float*)d_in[0];   // [T, H] fp32
    const float* W = (const float*)d_in[1];   // [E, H] fp32

    const int H = in_sizes[1] / NUM_EXPERTS;  // 2048
    const int T = in_sizes[0] / H;            // 16384

    float* out     = (float*)d_out;
    float* scores  = out;                     // T*2
    float* idx_out = out + (size_t)2 * T;     // T*2
    float* counts  = out + (size_t)4 * T;     // 8

    router_zero_counts<<<1, 32, 0, stream>>>(counts);

    const int blocks = T / TOKENS_PER_BLOCK;  // 128
    topk_router_wmma<<<blocks, THREADS, 0, stream>>>(x, W, scores, idx_out, counts, H);
}